// GATConv_14173392077060
// MI455X (gfx1250) — compile-verified
//
#include <hip/hip_runtime.h>
#include <hip/hip_bf16.h>

// GAT layer for MI455X (gfx1250, wave32).
// Phase 1 GEMM: V_WMMA_F32_16X16X4_F32 with double-buffered async
// global->LDS staging (GLOBAL_LOAD_ASYNC_TO_LDS_B128 / ASYNCcnt) when the
// toolchain exposes the builtin; synchronous LDS staging otherwise.
// Phases 2-5 are memory/atomic bound scatter-gather; per-node state
// (el/er/m/s = 6.4MB) lives in the 192MB L2 so edge atomics stay on-chip.

typedef __attribute__((ext_vector_type(2))) float v2f;
typedef __attribute__((ext_vector_type(8))) float v8f;
typedef __attribute__((ext_vector_type(4))) int   v4i;

#define BM 128
#define BN 32
#define BK 32
// Row stride 36 floats: row bases 144B -> 16B aligned (needed for B128 LDS
// writes), and 36*r mod 64 hits 16 distinct banks for r=0..15, so the
// strided A-fragment reads stay conflict-free.
#define LDA 36
#define LDB 36

#if __has_builtin(__builtin_amdgcn_global_load_async_to_lds_b128)
#define HAVE_ASYNC_LDS 1
#else
#define HAVE_ASYNC_LDS 0
#endif

__device__ __forceinline__ void stage16(const float* __restrict__ g, float* l) {
#if HAVE_ASYNC_LDS
    // Builtin signature (probe-confirmed via diagnostic):
    // (v4i addrspace(1)*, v4i addrspace(3)*, imm offset, imm cpol)
    __builtin_amdgcn_global_load_async_to_lds_b128(
        (__attribute__((address_space(1))) v4i*)(g),
        (__attribute__((address_space(3))) v4i*)(l),
        0, 0);
#else
    *(float4*)l = *(const float4*)g;
#endif
}

__device__ __forceinline__ void wait_stage() {
#if HAVE_ASYNC_LDS
#if __has_builtin(__builtin_amdgcn_s_wait_asynccnt)
    __builtin_amdgcn_s_wait_asynccnt(0);
#else
    asm volatile("s_wait_asynccnt 0x0" ::: "memory");
#endif
#endif
}

// ---------------------------------------------------------------------------
// Kernel 1: ft = feat @ W   (N x 256) = (N x 256)(256 x 256), f32 WMMA
// grid = (ceil(N/128), 256/32), block = 256 (8 waves)
// Each wave computes a 16x32 strip: two 16x16 WMMA accumulators sharing A.
// Double-buffered K-chunks: chunk c+1 streams into LDS via the async path
// while chunk c feeds the WMMAs.
// ---------------------------------------------------------------------------
__global__ __launch_bounds__(256) void gat_gemm(const float* __restrict__ feat,
                                                const float* __restrict__ W,
                                                float* __restrict__ ft,
                                                int N) {
    __shared__ float sA[2][BM * LDA];
    __shared__ float sB[2][BK * LDB];

    const int tid  = threadIdx.x;
    const int wave = tid >> 5;
    const int lane = tid & 31;
    const int lm   = lane & 15;              // col (B/C/D) or row (A) within tile
    const int khalf = (lane >> 4) ? 2 : 0;   // A fragment: lanes 16-31 hold K=2,3

    const int rowTile = blockIdx.x * BM;
    const int colBase = blockIdx.y * BN;

    v8f acc0 = {};
    v8f acc1 = {};

    // Stage one K-chunk (BM x BK of feat, BK x BN of W) into buffer `buf`.
    // Rows beyond N are left unstaged (garbage); their D rows are discarded
    // at the bounds-checked store, and A-row garbage only affects those rows.
    auto prefetch = [&](int buf, int k0) {
        #pragma unroll
        for (int i = 0; i < 4; ++i) {
            int idx = tid + i * 256;                 // 0..1023
            int r   = idx >> 3;
            int c4  = (idx & 7) * 4;
            int gr  = rowTile + r;
            if (gr < N)
                stage16(feat + (long long)gr * 256 + k0 + c4,
                        &sA[buf][r * LDA + c4]);
        }
        {
            int r  = tid >> 3;                       // 0..31
            int c4 = (tid & 7) * 4;
            stage16(W + (long long)(k0 + r) * 256 + colBase + c4,
                    &sB[buf][r * LDB + c4]);
        }
    };

    prefetch(0, 0);

    for (int c = 0; c < 256 / BK; ++c) {
        const int cur = c & 1;
        wait_stage();            // this wave's stages for chunk c have landed
        __syncthreads();         // everyone's stages for chunk c visible
        if (c + 1 < 256 / BK)
            prefetch(cur ^ 1, (c + 1) * BK);

        const float* pa = &sA[cur][(wave * 16 + lm) * LDA];
        const float* pb = &sB[cur][0];
        #pragma unroll
        for (int kk = 0; kk < BK; kk += 4) {
            const int kb = kk + khalf;
            v2f a;  a.x  = pa[kb];                 a.y  = pa[kb + 1];
            v2f b0; b0.x = pb[kb * LDB + lm];      b0.y = pb[(kb + 1) * LDB + lm];
            v2f b1; b1.x = pb[kb * LDB + 16 + lm]; b1.y = pb[(kb + 1) * LDB + 16 + lm];
            acc0 = __builtin_amdgcn_wmma_f32_16x16x4_f32(false, a, false, b0,
                                                         (short)0, acc0, false, false);
            acc1 = __builtin_amdgcn_wmma_f32_16x16x4_f32(false, a, false, b1,
                                                         (short)0, acc1, false, false);
        }
        __syncthreads();         // all waves done reading `cur` before reuse
    }

    // D layout: VGPR j -> row j (lanes 0-15) / row j+8 (lanes 16-31), col = lm.
    const int rb = rowTile + wave * 16 + (lane >> 4) * 8;
    #pragma unroll
    for (int j = 0; j < 8; ++j) {
        int gr = rb + j;
        if (gr < N) {
            ft[(long long)gr * 256 + colBase + lm]      = acc0[j];
            ft[(long long)gr * 256 + colBase + 16 + lm] = acc1[j];
        }
    }
}

// ---------------------------------------------------------------------------
// Kernel 2: per-node logits el/er, init m (-inf key) and s (0), out = bias.
// One block per node; head h = wave id, d = lane (D == 32 == wave32).
// ---------------------------------------------------------------------------
__global__ __launch_bounds__(256) void gat_logits_init(const float* __restrict__ ft,
                                                       const float* __restrict__ al,
                                                       const float* __restrict__ ar,
                                                       const float* __restrict__ bias,
                                                       float* __restrict__ el,
                                                       float* __restrict__ er,
                                                       unsigned* __restrict__ m,
                                                       float* __restrict__ s,
                                                       float* __restrict__ out) {
    const int n = blockIdx.x;
    const int t = threadIdx.x;
    const float v = ft[(long long)n * 256 + t];
    out[(long long)n * 256 + t] = bias[t];

    float pl = v * al[t];
    float pr = v * ar[t];
    #pragma unroll
    for (int o = 16; o > 0; o >>= 1) {
        pl += __shfl_down(pl, o, 32);
        pr += __shfl_down(pr, o, 32);
    }
    if ((t & 31) == 0) {
        const int h = t >> 5;
        el[n * 8 + h] = pl;
        er[n * 8 + h] = pr;
        m[n * 8 + h]  = 0x007FFFFFu;  // key(-inf): ~0xFF800000
        s[n * 8 + h]  = 0.f;
    }
}

// Monotonic float<->uint key for atomic max over IEEE floats.
__device__ __forceinline__ unsigned fkey(float f) {
    int i = __float_as_int(f);
    return (i >= 0) ? ((unsigned)i ^ 0x80000000u) : ~(unsigned)i;
}
__device__ __forceinline__ float funkey(unsigned u) {
    return (u & 0x80000000u) ? __uint_as_float(u ^ 0x80000000u)
                             : __uint_as_float(~u);
}

// ---------------------------------------------------------------------------
// Kernel 3: edge logits e = leakyrelu(el[src] + er[dst]); segment max via
// global_atomic_max_u32 on monotonic keys. One thread per (edge, head).
// ---------------------------------------------------------------------------
__global__ __launch_bounds__(256) void gat_edge_logit_max(const float* __restrict__ el,
                                                          const float* __restrict__ er,
                                                          const int* __restrict__ src,
                                                          const int* __restrict__ dst,
                                                          float* __restrict__ evals,
                                                          unsigned* __restrict__ m,
                                                          long long EH) {
    long long i = (long long)blockIdx.x * blockDim.x + threadIdx.x;
    if (i >= EH) return;
    const int e = (int)(i >> 3);
    const int h = (int)(i & 7);
    float v = el[(long long)src[e] * 8 + h] + er[(long long)dst[e] * 8 + h];
    v = (v > 0.f) ? v : 0.2f * v;
    evals[i] = v;
    atomicMax(&m[(long long)dst[e] * 8 + h], fkey(v));
}

// ---------------------------------------------------------------------------
// Kernel 4: ex = exp(e - m[dst]); segment sum via global_atomic_add_f32.
// ---------------------------------------------------------------------------
__global__ __launch_bounds__(256) void gat_edge_exp_sum(const int* __restrict__ dst,
                                                        float* __restrict__ evals,
                                                        const unsigned* __restrict__ m,
                                                        float* __restrict__ s,
                                                        long long EH) {
    long long i = (long long)blockIdx.x * blockDim.x + threadIdx.x;
    if (i >= EH) return;
    const int e = (int)(i >> 3);
    const int h = (int)(i & 7);
    const long long di = (long long)dst[e] * 8 + h;
    float ex = __expf(evals[i] - funkey(m[di]));
    evals[i] = ex;
    atomicAdd(&s[di], ex);
}

// ---------------------------------------------------------------------------
// Kernel 5: out[dst] += ft[src] * (ex / s[dst]).  64 threads/edge, each
// owning a contiguous float4 of the 256 channels (head h = chunk/8).
// ---------------------------------------------------------------------------
__global__ __launch_bounds__(256) void gat_aggregate(const float* __restrict__ ft,
                                                     const float* __restrict__ evals,
                                                     const float* __restrict__ s,
                                                     const int* __restrict__ src,
                                                     const int* __restrict__ dst,
                                                     float* __restrict__ out,
                                                     int E) {
    long long gid = (long long)blockIdx.x * blockDim.x + threadIdx.x;
    const int e = (int)(gid >> 6);
    if (e >= E) return;
    const int t = (int)(gid & 63);  // channel chunk: channels [t*4, t*4+4)
    const int h = t >> 3;
    const int se = src[e];
    const int de = dst[e];
    const float a = evals[(long long)e * 8 + h] / s[(long long)de * 8 + h];
    const float4 f = *(const float4*)(ft + (long long)se * 256 + t * 4);
    float* o = out + (long long)de * 256 + t * 4;
    atomicAdd(o + 0, f.x * a);
    atomicAdd(o + 1, f.y * a);
    atomicAdd(o + 2, f.z * a);
    atomicAdd(o + 3, f.w * a);
}

// ---------------------------------------------------------------------------
extern "C" void kernel_launch(void* const* d_in, const int* in_sizes, int n_in,
                              void* d_out, int out_size, void* d_ws, size_t ws_size,
                              hipStream_t stream) {
    const float* feat = (const float*)d_in[0];
    const float* W    = (const float*)d_in[1];
    const float* al   = (const float*)d_in[2];
    const float* ar   = (const float*)d_in[3];
    const float* bias = (const float*)d_in[4];
    const int*   src  = (const int*)d_in[5];
    const int*   dst  = (const int*)d_in[6];

    const int Din = 256, HD = 256, H = 8;
    const int N = in_sizes[0] / Din;
    const long long E = in_sizes[5];
    float* out = (float*)d_out;

    // Workspace carve-up (~109 MB): ft | el | er | m | s | ex
    char* w = (char*)d_ws;
    float*    ft = (float*)w;     w += (size_t)N * HD * sizeof(float);
    float*    el = (float*)w;     w += (size_t)N * H * sizeof(float);
    float*    er = (float*)w;     w += (size_t)N * H * sizeof(float);
    unsigned* m  = (unsigned*)w;  w += (size_t)N * H * sizeof(unsigned);
    float*    s  = (float*)w;     w += (size_t)N * H * sizeof(float);
    float*    ex = (float*)w;

    // 1) projection GEMM (WMMA f32, async double-buffered staging)
    dim3 ggrid((N + BM - 1) / BM, HD / BN);
    gat_gemm<<<ggrid, 256, 0, stream>>>(feat, W, ft, N);

    // 2) node logits + state init + out = bias
    gat_logits_init<<<N, 256, 0, stream>>>(ft, al, ar, bias, el, er, m, s, out);

    // 3) edge logits + segment max
    const long long EH = E * H;
    const int eb = (int)((EH + 255) / 256);
    gat_edge_logit_max<<<eb, 256, 0, stream>>>(el, er, src, dst, ex, m, EH);

    // 4) exp + segment sum
    gat_edge_exp_sum<<<eb, 256, 0, stream>>>(dst, ex, m, s, EH);

    // 5) weighted scatter-aggregate
    const long long tot = E * 64;
    gat_aggregate<<<(int)((tot + 255) / 256), 256, 0, stream>>>(ft, ex, s, src, dst, out, (int)E);
}